// Newmark_beta_solver_layer_40286793236805
// MI455X (gfx1250) — compile-verified
//
#include <hip/hip_runtime.h>

typedef __attribute__((ext_vector_type(2))) float v2f;
typedef __attribute__((ext_vector_type(8))) float v8f;

#define B_   64
#define T_   8192
#define M_   8
#define D_   128
#define L_   128          // time-block length
#define NB_  64           // T_/L_
#define EPS_ 1e-12

struct Coef { double dt, p1, p2, p3, p4, invL, cc, kk; };

__device__ inline Coef make_coef(double dt, double mm, double cc, double kk) {
    Coef C;
    C.dt = dt;
    C.p1 = 0.5 * dt;            // (1-gamma)*dt
    C.p2 = 0.25 * dt * dt;      // (0.5-beta)*dt^2
    C.p3 = 0.5 * dt;            // gamma*dt
    C.p4 = 0.25 * dt * dt;      // beta*dt^2
    double LHS = mm + 0.5 * dt * cc + 0.25 * dt * dt * kk;
    C.invL = 1.0 / (LHS + EPS_);
    C.cc = cc; C.kk = kk;
    return C;
}

__device__ inline void nb_step(const Coef& C, double Q,
                               double& q, double& qd, double& qdd) {
    double qp  = q  + C.dt * qd + C.p2 * qdd;
    double qdp = qd + C.p1 * qdd;
    double rhs = Q - C.cc * qdp - C.kk * qp;
    double a   = rhs * C.invL;
    qd  = qdp + C.p3 * a;
    q   = qp  + C.p4 * a;
    qdd = a;
}

__device__ inline void mat3_mul(const double X[3][3], const double Y[3][3],
                                double Z[3][3]) {
    for (int r = 0; r < 3; ++r)
        for (int c = 0; c < 3; ++c)
            Z[r][c] = X[r][0]*Y[0][c] + X[r][1]*Y[1][c] + X[r][2]*Y[2][c];
}

// --- Kernel 1: per-(chain, time-block) zero-state aggregates (parallel) -----
__global__ void k1_block_aggregate(const float* __restrict__ Q,
                                   const float* __restrict__ tvec,
                                   const float* __restrict__ mcol,
                                   const float* __restrict__ ccol,
                                   const float* __restrict__ kcol,
                                   double* __restrict__ rbuf) {
    int gid   = blockIdx.x * blockDim.x + threadIdx.x;   // 0..32767
    int j     = gid & (NB_ - 1);
    int chain = gid >> 6;
    int b = chain >> 3, m = chain & 7;

    double dt = (double)tvec[b * T_ + 1] - (double)tvec[b * T_];
    Coef C = make_coef(dt, (double)mcol[m], (double)ccol[m], (double)kcol[m]);

    const float* Qb = Q + (size_t)b * T_ * M_ + m;
    double q = 0.0, qd = 0.0, qdd = 0.0;
    int t0 = j * L_, t1 = t0 + L_;
    if (j == 0) {   // absolute aggregate for block 0: includes true S_0
        qdd = (double)Qb[0] / ((double)mcol[m] + EPS_);
        t0 = 1;
    }
    for (int t = t0; t < t1; ++t)
        nb_step(C, (double)Qb[(size_t)t * M_], q, qd, qdd);

    double* r = rbuf + (size_t)(chain * NB_ + j) * 3;
    r[0] = q; r[1] = qd; r[2] = qdd;
}

// --- Kernel 2: per-chain scan of block aggregates (A^128 via squaring) ------
__global__ void k2_scan(const float* __restrict__ tvec,
                        const float* __restrict__ mcol,
                        const float* __restrict__ ccol,
                        const float* __restrict__ kcol,
                        const double* __restrict__ rbuf,
                        double* __restrict__ Ebuf) {
    int chain = blockIdx.x * blockDim.x + threadIdx.x;
    if (chain >= B_ * M_) return;
    int b = chain >> 3, m = chain & 7;

    double dt = (double)tvec[b * T_ + 1] - (double)tvec[b * T_];
    Coef C = make_coef(dt, (double)mcol[m], (double)ccol[m], (double)kcol[m]);

    // Build A: columns = step applied to basis states with zero forcing.
    double A[3][3];
    for (int col = 0; col < 3; ++col) {
        double q = (col == 0), qd = (col == 1), qdd = (col == 2);
        nb_step(C, 0.0, q, qd, qdd);
        A[0][col] = q; A[1][col] = qd; A[2][col] = qdd;
    }
    // P = A^128 (7 squarings)
    double P[3][3], Pt[3][3];
    for (int r = 0; r < 3; ++r) for (int c = 0; c < 3; ++c) P[r][c] = A[r][c];
    for (int s = 0; s < 7; ++s) {
        mat3_mul(P, P, Pt);
        for (int r = 0; r < 3; ++r) for (int c = 0; c < 3; ++c) P[r][c] = Pt[r][c];
    }

    const double* r0 = rbuf + (size_t)chain * NB_ * 3;
    double*       E  = Ebuf + (size_t)chain * NB_ * 3;

    double S[3] = { r0[0], r0[1], r0[2] };           // absolute S_{L-1}
    E[3 + 0] = S[0]; E[3 + 1] = S[1]; E[3 + 2] = S[2];   // E_1
    for (int j = 1; j < NB_ - 1; ++j) {
        const double* rj = r0 + (size_t)j * 3;
        double ns[3];
        for (int r = 0; r < 3; ++r)
            ns[r] = P[r][0]*S[0] + P[r][1]*S[1] + P[r][2]*S[2] + rj[r];
        S[0] = ns[0]; S[1] = ns[1]; S[2] = ns[2];
        double* Ej = E + (size_t)(j + 1) * 3;
        Ej[0] = S[0]; Ej[1] = S[1]; Ej[2] = S[2];
    }
}

// --- Kernel 3: replay block + fused WMMA GEMM (Phi @ qdd) -------------------
__global__ __launch_bounds__(256)
void k3_replay_gemm(const float* __restrict__ Q,
                    const float* __restrict__ tvec,
                    const float* __restrict__ Phi,
                    const float* __restrict__ mcol,
                    const float* __restrict__ ccol,
                    const float* __restrict__ kcol,
                    const double* __restrict__ Ebuf,
                    float* __restrict__ out) {
    __shared__ float qdd_s[L_ * 9];                 // stride 9: conflict-free
    __shared__ __align__(16) float out_s[16 * 140]; // stride 140: conflict-free, 16B rows

    int j = blockIdx.x, b = blockIdx.y;
    int tid = threadIdx.x;

    // Phase A: 8 threads replay 128 steps in f64 (block-entry state from scan)
    if (tid < M_) {
        int m = tid, chain = b * M_ + m;
        double dt = (double)tvec[b * T_ + 1] - (double)tvec[b * T_];
        Coef C = make_coef(dt, (double)mcol[m], (double)ccol[m], (double)kcol[m]);
        const float* Qb = Q + (size_t)b * T_ * M_ + m;
        double q, qd, qdd;
        if (j == 0) {
            q = 0.0; qd = 0.0;
            qdd = (double)Qb[0] / ((double)mcol[m] + EPS_);
            qdd_s[0 * 9 + m] = (float)qdd;
            for (int t = 1; t < L_; ++t) {
                nb_step(C, (double)Qb[(size_t)t * M_], q, qd, qdd);
                qdd_s[t * 9 + m] = (float)qdd;
            }
        } else {
            const double* E = Ebuf + (size_t)(chain * NB_ + j) * 3;
            q = E[0]; qd = E[1]; qdd = E[2];
            int t0 = j * L_;
            for (int tl = 0; tl < L_; ++tl) {
                nb_step(C, (double)Qb[(size_t)(t0 + tl) * M_], q, qd, qdd);
                qdd_s[tl * 9 + m] = (float)qdd;
            }
        }
    }
    __syncthreads();

    // Phase B: WMMA GEMM out[t,d] = sum_m qdd[t,m]*Phi[d,m]
    int wave = tid >> 5, lane = tid & 31;
    int kp = (lane >> 4) * 2;       // K sub-pair per half-wave (A/B layout)
    int n  = lane & 15;             // N (=d) / row (=t) index within tile
    int dbase = wave * 16, d = dbase + n;

    // B operand (Phi^T tiles), fixed per wave: lane -> B[k, n] = Phi[d, k]
    v2f bA = { Phi[d * M_ + kp],     Phi[d * M_ + kp + 1]     }; // K=0..3
    v2f bB = { Phi[d * M_ + 4 + kp], Phi[d * M_ + 4 + kp + 1] }; // K=4..7

    for (int s = 0; s < 8; ++s) {
        int tb = s * 16;
        int tl = tb + n;            // A row (= t) for this lane
        v2f a0 = { qdd_s[tl * 9 + kp],     qdd_s[tl * 9 + kp + 1]     };
        v2f a1 = { qdd_s[tl * 9 + 4 + kp], qdd_s[tl * 9 + 4 + kp + 1] };

        v8f c = {};
        c = __builtin_amdgcn_wmma_f32_16x16x4_f32(false, a0, false, bA,
                                                  (short)0, c, false, false);
        c = __builtin_amdgcn_wmma_f32_16x16x4_f32(false, a1, false, bB,
                                                  (short)0, c, false, false);

        // C tile (16 t x 16 d) -> LDS staging
        int thalf = (lane >> 4) * 8;
#pragma unroll
        for (int r = 0; r < 8; ++r)
            out_s[(r + thalf) * 140 + d] = c[r];
        __syncthreads();

        // Coalesced float4 store: 16 rows x 128 contiguous floats
        size_t base = ((size_t)b * T_ + (size_t)j * L_ + tb) * D_;
        for (int f = tid; f < 16 * 32; f += 256) {
            int row = f >> 5, c4 = f & 31;
            float4 v = *(const float4*)&out_s[row * 140 + c4 * 4];
            *(float4*)(out + base + (size_t)row * D_ + (size_t)c4 * 4) = v;
        }
        __syncthreads();
    }
}

extern "C" void kernel_launch(void* const* d_in, const int* in_sizes, int n_in,
                              void* d_out, int out_size, void* d_ws, size_t ws_size,
                              hipStream_t stream) {
    const float* Qpred = (const float*)d_in[0];
    const float* tvec  = (const float*)d_in[1];
    const float* Phi   = (const float*)d_in[2];
    const float* mcol  = (const float*)d_in[3];
    const float* ccol  = (const float*)d_in[4];
    const float* kcol  = (const float*)d_in[5];
    float* out = (float*)d_out;

    double* rbuf = (double*)d_ws;                       // 512*64*3 doubles
    double* Ebuf = rbuf + (size_t)B_ * M_ * NB_ * 3;    // 512*64*3 doubles

    k1_block_aggregate<<<(B_ * M_ * NB_) / 256, 256, 0, stream>>>(
        Qpred, tvec, mcol, ccol, kcol, rbuf);
    k2_scan<<<(B_ * M_ + 255) / 256, 256, 0, stream>>>(
        tvec, mcol, ccol, kcol, rbuf, Ebuf);
    dim3 g3(NB_, B_);
    k3_replay_gemm<<<g3, 256, 0, stream>>>(
        Qpred, tvec, Phi, mcol, ccol, kcol, Ebuf, out);
}